// MessagePassing_54820962566736
// MI455X (gfx1250) — compile-verified
//
#include <hip/hip_runtime.h>

// Scatter-add message passing: out[dst[e]] += x[src[e]], d=64 fp32.
// L2-resident problem (x/out/edges all << 192MB L2). Bound by L2 fp32-atomic
// throughput, so: coalesced 64B-per-halfwave loads+atomics, async-to-LDS
// double-buffered edge-index staging (CDNA5 ASYNCcnt path).

#define TILE  128   // edges staged per LDS buffer (== BLOCK/2, one 8B copy per thread)
#define BLOCK 256   // 8 wave32 per block
#define FEAT  64

typedef unsigned long long u64;

template<int N>
__device__ __forceinline__ void wait_async() {
#if __has_builtin(__builtin_amdgcn_s_wait_asynccnt)
    __builtin_amdgcn_s_wait_asynccnt(N);
#else
    asm volatile("s_wait_asynccnt %0" :: "i"(N) : "memory");
#endif
}

// Async copy 8B global -> LDS (GLOBAL_LOAD_ASYNC_TO_LDS_B64, GV mode).
// Low 32 bits of a flat shared-aperture pointer are the LDS byte offset
// (aperture lives in addr[63:32] on CDNA5).
__device__ __forceinline__ void async_cp_b64(void* lds_ptr, const void* gptr) {
    unsigned lds = (unsigned)(u64)lds_ptr;
    asm volatile("global_load_async_to_lds_b64 %0, %1, off"
                 :: "v"(lds), "v"(gptr)
                 : "memory");
}

__global__ void zero_kernel(float* __restrict__ p, int n) {
    int i = blockIdx.x * blockDim.x + threadIdx.x;
    int b = i * 4;
    if (b + 3 < n) {
        *(float4*)(p + b) = make_float4(0.f, 0.f, 0.f, 0.f);
    } else {
        for (int j = 0; j < 4; ++j)
            if (b + j < n) p[b + j] = 0.f;
    }
}

// Stage one 128-edge tile: threads 0..127 copy src indices, 128..255 dst.
__device__ __forceinline__ void stage_tile(const u64* __restrict__ srcA,
                                           const u64* __restrict__ dstA,
                                           u64 (*sb)[TILE], int tbase, int E,
                                           int tid) {
    int k  = tid & (TILE - 1);
    int gi = tbase + k;
    if (gi >= E) gi = E - 1;                      // clamp: stays in-bounds, never processed
    const u64* g = (tid < TILE) ? (srcA + gi) : (dstA + gi);
    async_cp_b64(&sb[tid >> 7][k], g);
}

__global__ __launch_bounds__(BLOCK) void scatter_add_kernel(
        const float* __restrict__ x,
        const u64*  __restrict__ srcA,
        const u64*  __restrict__ dstA,
        float*      __restrict__ out,
        int E, int ntiles) {
    static_assert(TILE * 2 == BLOCK, "one 8B async copy per thread");
    __shared__ u64 sbuf[2][2][TILE];              // [buffer][src/dst][edge]

    const int tid  = threadIdx.x;
    const int slot = tid >> 4;                    // 0..15 : edge within a pass
    const int c    = tid & 15;                    // 0..15 : feature lane

    int tile = blockIdx.x;
    int buf  = 0;
    if (tile < ntiles)
        stage_tile(srcA, dstA, sbuf[0], tile * TILE, E, tid);

    for (; tile < ntiles; tile += gridDim.x) {
        const int nxt = tile + gridDim.x;
        if (nxt < ntiles) {
            stage_tile(srcA, dstA, sbuf[buf ^ 1], nxt * TILE, E, tid);
            wait_async<1>();                      // tile `tile` done, next in flight
        } else {
            wait_async<0>();
        }
        __syncthreads();                          // staging visible to all waves

        const int base = tile * TILE;
#pragma unroll
        for (int p = 0; p < TILE / 16; ++p) {     // 16 edges per pass over the block
            const int sl = p * 16 + slot;
            if (base + sl < E) {
                const int s = (int)sbuf[buf][0][sl];
                const int d = (int)sbuf[buf][1][sl];
                const float* xr   = x   + (size_t)s * FEAT;
                float*       orow = out + (size_t)d * FEAT;
                // lanes c=0..15 -> 64B contiguous per load / per atomic instruction
                float v0 = xr[c];
                float v1 = xr[c + 16];
                float v2 = xr[c + 32];
                float v3 = xr[c + 48];
                atomicAdd(orow + c,      v0);
                atomicAdd(orow + c + 16, v1);
                atomicAdd(orow + c + 32, v2);
                atomicAdd(orow + c + 48, v3);
            }
        }
        __syncthreads();                          // done reading buf before it is re-staged
        buf ^= 1;
    }
}

extern "C" void kernel_launch(void* const* d_in, const int* in_sizes, int n_in,
                              void* d_out, int out_size, void* d_ws, size_t ws_size,
                              hipStream_t stream) {
    const float* x  = (const float*)d_in[0];
    const u64*   ei = (const u64*)d_in[1];        // int64 [2, E]: row0=src, row1=dst
    const int    E  = in_sizes[1] / 2;
    const u64* srcA = ei;
    const u64* dstA = ei + E;
    float* out = (float*)d_out;

    // out must be zero before atomic accumulation, every call.
    int nz = (out_size + 4 * 256 - 1) / (4 * 256);
    if (nz > 0) zero_kernel<<<nz, 256, 0, stream>>>(out, out_size);

    int ntiles = (E + TILE - 1) / TILE;
    if (ntiles <= 0) return;
    int grid = ntiles < 2048 ? ntiles : 2048;
    scatter_add_kernel<<<grid, BLOCK, 0, stream>>>(x, srcA, dstA, out, E, ntiles);
}